// hierarch_loss_730144440410
// MI455X (gfx1250) — compile-verified
//
#include <hip/hip_runtime.h>
#include <hip/hip_bf16.h>
#include <stdint.h>

// ---------------------------------------------------------------------------
// hierarch-loss pairwise ranking penalty for MI455X (gfx1250, wave32)
//
//   out = ALPHA * sum over events/pairs of masked exp(+/- diff / SIGMA)
//
// Compute-bound: 3*8192^2 ~= 2e8 pair evaluations, each ~15 VALU + <=2 v_exp.
// The 9.8MB cumT table lives in L2 (192MB); per-event (t,lab) table is staged
// into LDS with the CDNA5 Tensor Data Mover (tensor_load_to_lds + TENSORcnt).
// Block reductions use V_WMMA_F32_16X16X4_F32 (exact f32, fixed order):
// with A = ones(16x4), D[m,n] = sum_k B[k,n] + C[m,n]; summing the 16 column
// sums of C yields the total regardless of how values map into B slots.
// ---------------------------------------------------------------------------

#define SIGMA_INV 10.0f
#define ALPHA_C   0.05f
#define BPE       256      // blocks per event
#define BLOCK     256      // threads per block (8 wave32)
#define S_MAX     8192

typedef uint32_t u32;
typedef uint64_t u64;
typedef u32   v4u __attribute__((ext_vector_type(4)));
typedef int   v4i __attribute__((ext_vector_type(4)));
typedef int   v8i __attribute__((ext_vector_type(8)));
typedef float v2f __attribute__((ext_vector_type(2)));
typedef float v8f __attribute__((ext_vector_type(8)));

// --- Tensor Data Mover: 1-D dword copy  global -> LDS  (D# per CDNA5 ISA §8)
__device__ __forceinline__ void tdm_load_1d_to_lds(void* lds_dst,
                                                   const void* gsrc,
                                                   u32 ndwords) {
  u64 ga  = (u64)(uintptr_t)gsrc;
  // generic shared pointer = {shared_aperture_hi32, lds_offset_lo32}
  u32 lds = (u32)(uintptr_t)lds_dst;

  v4u g0 = {0u, 0u, 0u, 0u};
  g0.x = 1u;                                         // count=1 (valid), user mode
  g0.y = lds;                                        // lds_addr
  g0.z = (u32)ga;                                    // global_addr[31:0]
  g0.w = ((u32)(ga >> 32) & 0x01FFFFFFu)             // global_addr[56:32]
         | (2u << 30);                               // type = 2 ("image")

  v8i g1 = {0, 0, 0, 0, 0, 0, 0, 0};
  g1[0] = (int)(2u << 16);                           // wg_mask=0, data_size=2 (4B)
  g1[1] = (int)((ndwords & 0xFFFFu) << 16);          // tensor_dim0[15:0]
  g1[2] = (int)((ndwords >> 16) & 0xFFFFu);          // tensor_dim0[31:16], dim1=0
  g1[3] = (int)((ndwords & 0xFFFFu) << 16);          // tile_dim0 (16-bit)
  g1[5] = (int)ndwords;                              // tensor_dim0_stride[31:0]

  v4i z4 = {0, 0, 0, 0};
#if __has_include(<hip/amd_detail/amd_gfx1250_TDM.h>)
  v8i z8 = {0, 0, 0, 0, 0, 0, 0, 0};
  __builtin_amdgcn_tensor_load_to_lds(g0, g1, z4, z4, z8, 0);   // clang-23 form
#else
  __builtin_amdgcn_tensor_load_to_lds(g0, g1, z4, z4, 0);       // ROCm 7.2 form
#endif
}

// --- WMMA-based sum of s_red[0..255]; call from wave 0 (all 32 lanes active).
// Returns the total in lane 0 (the xor butterfly leaves it in every lane).
__device__ __forceinline__ float wmma_sum256(const float* s_red, int lane) {
  v2f ones; ones.x = 1.0f; ones.y = 1.0f;            // A = ones(16x4)
  v8f c = {0.f, 0.f, 0.f, 0.f, 0.f, 0.f, 0.f, 0.f};
#pragma unroll
  for (int j = 0; j < 4; ++j) {                      // 4 x 64-value chunks
    v2f b;
    b.x = s_red[j * 64 + lane];
    b.y = s_red[j * 64 + 32 + lane];
    // v_wmma_f32_16x16x4_f32: D = A*B + C  (exact f32 FMA, RNE)
    c = __builtin_amdgcn_wmma_f32_16x16x4_f32(
        /*neg_a=*/false, ones, /*neg_b=*/false, b,
        /*c_mod=*/(short)0, c, /*reuse_a=*/false, /*reuse_b=*/false);
  }
  // C row M=0 (VGPR0, lanes 0..15) = the 16 column sums; butterfly low 16.
  float x = c[0];
  x += __shfl_xor(x, 8, 32);
  x += __shfl_xor(x, 4, 32);
  x += __shfl_xor(x, 2, 32);
  x += __shfl_xor(x, 1, 32);
  return x;                                          // lane 0: grand total
}

// --- Kernel 1: running cumsum over T, stored transposed; pack (t, lab) -------
__global__ void prep_kernel(const float* __restrict__ ev,
                            const int* __restrict__ times,
                            const int* __restrict__ labs,
                            float* __restrict__ cumT,   // [E][T][S]
                            int* __restrict__ tlpack,   // [E][S]
                            int S, int T) {
  const int e = blockIdx.y;
  const int s = blockIdx.x * blockDim.x + threadIdx.x;
  if (s >= S) return;

  const float* row = ev + ((size_t)e * S + s) * (size_t)T;
  float run = 0.0f;
  for (int k = 0; k < T; ++k) {
    run += row[k];
    cumT[((size_t)e * T + k) * (size_t)S + s] = run;   // coalesced over s
  }
  const int E = gridDim.y;
  int t = times[(size_t)s * E + e];
  int l = labs[(size_t)s * E + e];
  tlpack[(size_t)e * S + s] = (t & 0xFFFF) | (l << 16);
}

// --- Kernel 2: pairwise penalty; rows are wave-uniform, lanes sweep b --------
__global__ void penalty_kernel(const float* __restrict__ cumT,
                               const int* __restrict__ tlpack,
                               float* __restrict__ partials,
                               int S, int T) {
  __shared__ int   s_tl[S_MAX];
  __shared__ float s_red[BLOCK];

  const int e   = blockIdx.y;
  const int blk = blockIdx.x;

  // Stage this event's packed (t,lab) table (32KB) into LDS via the TDM.
  if (threadIdx.x == 0) {
    tdm_load_1d_to_lds(s_tl, tlpack + (size_t)e * S, (u32)S);
    __builtin_amdgcn_s_wait_tensorcnt(0);
  }
  __syncthreads();

  const float* cume = cumT + (size_t)e * T * S;
  const int rows = (S + BPE - 1) / BPE;
  const int a0   = blk * rows;

  float acc = 0.0f;
  for (int r = 0; r < rows; ++r) {
    int a = a0 + r;
    if (a >= S) break;

    // Row-level constants: identical across the wave -> scalar branches.
    int tla   = s_tl[a];
    int t_a   = tla & 0xFFFF;
    int lab_a = tla >> 16;
    bool doF  = (lab_a == 1);
    bool doB  = (t_a > 0);
    if (!doF && !doB) continue;

    int e_f = min(t_a + 1, T - 1);
    int k_f = e_f - 1;
    int e_b = min(doF ? t_a : t_a + 1, T - 1);
    int k_b = max(e_b, 1) - 1;

    const float* rowF = cume + (size_t)k_f * S;
    const float* rowB = cume + (size_t)k_b * S;
    __builtin_prefetch(rowF + threadIdx.x, 0, 0);      // global_prefetch_b8
    __builtin_prefetch(rowB + threadIdx.x, 0, 0);
    float c_af = rowF[a];                              // wave-uniform scalars
    float c_ab = rowB[a];

    for (int b = threadIdx.x; b < S; b += BLOCK) {
      int tlb   = s_tl[b];
      int t_b   = tlb & 0xFFFF;
      int lab_b = tlb >> 16;

      if (doF) {                                       // forward ranking term
        float cb  = rowF[b];                           // L2-hot, coalesced
        float arg = (cb - c_af) * SIGMA_INV;           // -diff_f/sigma
        bool cond = (t_b >= e_f) | ((t_b == e_f - 1) & (lab_b == 0));
        if (cond & (arg > 0.0f)) acc += __expf(arg);
      }
      if (doB) {                                       // backward ranking term
        float cb  = rowB[b];
        float arg = (c_ab - cb) * SIGMA_INV;           //  diff_b/sigma
        bool cond = (t_b <= e_b - 1) & (lab_b == 1);
        if (cond & (arg > 0.0f)) acc += __expf(arg);
      }
    }
  }

  // Deterministic block reduction on the matrix pipe (f32 WMMA).
  s_red[threadIdx.x] = acc;
  __syncthreads();
  if (threadIdx.x < 32) {
    float tot = wmma_sum256(s_red, threadIdx.x);
    if (threadIdx.x == 0) partials[(size_t)e * BPE + blk] = tot;
  }
}

// --- Kernel 3: fixed-order final reduction (WMMA again) ----------------------
__global__ void finalize_kernel(const float* __restrict__ partials, int n,
                                float* __restrict__ out) {
  __shared__ float s_red[BLOCK];
  float acc = 0.0f;
  for (int i = threadIdx.x; i < n; i += BLOCK) acc += partials[i];
  s_red[threadIdx.x] = acc;
  __syncthreads();
  if (threadIdx.x < 32) {
    float tot = wmma_sum256(s_red, threadIdx.x);
    if (threadIdx.x == 0) out[0] = ALPHA_C * tot;
  }
}

// ---------------------------------------------------------------------------
extern "C" void kernel_launch(void* const* d_in, const int* in_sizes, int n_in,
                              void* d_out, int out_size, void* d_ws, size_t ws_size,
                              hipStream_t stream) {
  const float* ev    = (const float*)d_in[0];   // (E,S,T) f32
  const int*   times = (const int*)d_in[1];     // (S,E) i32
  const int*   labs  = (const int*)d_in[2];     // (S,E) i32

  const int E = 3;
  const int SE = in_sizes[1];                   // S*E
  const int S  = SE / E;                        // 8192
  const int T  = in_sizes[0] / SE;              // 100

  // Workspace layout: cumT[E][T][S] f32 | tlpack[E][S] i32 | partials[E*BPE] f32
  char* ws = (char*)d_ws;
  float* cumT     = (float*)ws;
  int*   tlpack   = (int*)(ws + (size_t)E * T * S * sizeof(float));
  float* partials = (float*)(ws + (size_t)E * T * S * sizeof(float)
                                + (size_t)E * S * sizeof(int));

  prep_kernel<<<dim3((S + BLOCK - 1) / BLOCK, E), BLOCK, 0, stream>>>(
      ev, times, labs, cumT, tlpack, S, T);

  penalty_kernel<<<dim3(BPE, E), BLOCK, 0, stream>>>(cumT, tlpack, partials, S, T);

  finalize_kernel<<<1, BLOCK, 0, stream>>>(partials, E * BPE, (float*)d_out);
}